// DMI_50388556316704
// MI455X (gfx1250) — compile-verified
//
#include <hip/hip_runtime.h>
#include <stdint.h>

#define N 192
#define TILE 4
#define NROWS 11
#define RSTRIDE 200   // floats per LDS row (192 data + 2 pad + slack)

typedef __attribute__((ext_vector_type(2))) float v2f;
typedef __attribute__((ext_vector_type(8))) float v8f;

// Async copy: each active lane copies 4 bytes from the global address base plus voff to the LDS address in lds_addr.
// GVS addressing mode; tracked by ASYNCcnt.
__device__ __forceinline__ void async_row_load(uint32_t lds_addr, uint32_t voff,
                                               const float* __restrict__ base) {
    asm volatile("global_load_async_to_lds_b32 %0, %1, %2"
                 :: "v"(lds_addr), "v"(voff), "s"(base)
                 : "memory");
}

__global__ __launch_bounds__(192) void dmi_stencil_kernel(
    const float* __restrict__ x, float* __restrict__ partial)
{
    __shared__ float smem[NROWS * RSTRIDE];
    __shared__ float wsum[8];

    const int t   = threadIdx.x;          // a3 index, 0..191
    const int a1  = blockIdx.y;           // 0..191
    const int a20 = blockIdx.x * TILE;    // first a2 of this block's tile

    const uint32_t lds_base = (uint32_t)(uintptr_t)&smem[0]; // LDS byte offset = low 32 bits
    const uint32_t voff = (uint32_t)t * 4u;
    const uint32_t dst  = lds_base + (uint32_t)(1 + t) * 4u; // +1 = x-shift pad slot

    // Zero the x-direction pad cells of rows 0 (comp0) and 1 (comp1) once.
    if (t == 0) {
        smem[0 * RSTRIDE + 0]   = 0.0f; smem[0 * RSTRIDE + 193] = 0.0f;
        smem[1 * RSTRIDE + 0]   = 0.0f; smem[1 * RSTRIDE + 193] = 0.0f;
    }
    // a1-boundary neighbor rows: zero once (uniform per block; never async-written).
    if (a1 == 0)     { smem[3 * RSTRIDE + 1 + t] = 0.0f; smem[5 * RSTRIDE + 1 + t] = 0.0f; }
    if (a1 == N - 1) { smem[4 * RSTRIDE + 1 + t] = 0.0f; smem[6 * RSTRIDE + 1 + t] = 0.0f; }

    const size_t plane = (size_t)N * N;
    const size_t comp  = plane * N;

    float esum = 0.0f;

    for (int it = 0; it < TILE; ++it) {
        const int a2 = a20 + it;
        const float* r0 = x + 0 * comp + (size_t)a1 * plane + (size_t)a2 * N;
        const float* r1 = x + 1 * comp + (size_t)a1 * plane + (size_t)a2 * N;
        const float* r2 = x + 2 * comp + (size_t)a1 * plane + (size_t)a2 * N;

        // Stage all needed rows into LDS with async copies (uniform branches only).
        async_row_load(dst + 0u * RSTRIDE * 4u, voff, r0);
        async_row_load(dst + 1u * RSTRIDE * 4u, voff, r1);
        async_row_load(dst + 2u * RSTRIDE * 4u, voff, r2);
        if (a1 > 0) {
            async_row_load(dst + 3u * RSTRIDE * 4u, voff, r1 - plane);
            async_row_load(dst + 5u * RSTRIDE * 4u, voff, r2 - plane);
        }
        if (a1 < N - 1) {
            async_row_load(dst + 4u * RSTRIDE * 4u, voff, r1 + plane);
            async_row_load(dst + 6u * RSTRIDE * 4u, voff, r2 + plane);
        }
        if (a2 > 0) {
            async_row_load(dst + 7u * RSTRIDE * 4u, voff, r0 - N);
            async_row_load(dst + 9u * RSTRIDE * 4u, voff, r2 - N);
        } else {
            smem[7 * RSTRIDE + 1 + t] = 0.0f;
            smem[9 * RSTRIDE + 1 + t] = 0.0f;
        }
        if (a2 < N - 1) {
            async_row_load(dst +  8u * RSTRIDE * 4u, voff, r0 + N);
            async_row_load(dst + 10u * RSTRIDE * 4u, voff, r2 + N);
        } else {
            smem[ 8 * RSTRIDE + 1 + t] = 0.0f;
            smem[10 * RSTRIDE + 1 + t] = 0.0f;
        }

        asm volatile("s_wait_asynccnt 0" ::: "memory");
        __syncthreads();

        const float c0   = smem[ 0 * RSTRIDE + 1 + t];
        const float c1   = smem[ 1 * RSTRIDE + 1 + t];
        const float c2   = smem[ 2 * RSTRIDE + 1 + t];
        const float c1m1 = smem[ 3 * RSTRIDE + 1 + t];  // comp1 @ a1-1
        const float c1p1 = smem[ 4 * RSTRIDE + 1 + t];  // comp1 @ a1+1
        const float c2m1 = smem[ 5 * RSTRIDE + 1 + t];  // comp2 @ a1-1
        const float c2p1 = smem[ 6 * RSTRIDE + 1 + t];  // comp2 @ a1+1
        const float c0m2 = smem[ 7 * RSTRIDE + 1 + t];  // comp0 @ a2-1
        const float c0p2 = smem[ 8 * RSTRIDE + 1 + t];  // comp0 @ a2+1
        const float c2m2 = smem[ 9 * RSTRIDE + 1 + t];  // comp2 @ a2-1
        const float c2p2 = smem[10 * RSTRIDE + 1 + t];  // comp2 @ a2+1
        const float c0xm = smem[ 0 * RSTRIDE + 0 + t];  // comp0 @ a3-1
        const float c0xp = smem[ 0 * RSTRIDE + 2 + t];  // comp0 @ a3+1
        const float c1xm = smem[ 1 * RSTRIDE + 0 + t];  // comp1 @ a3-1
        const float c1xp = smem[ 1 * RSTRIDE + 2 + t];  // comp1 @ a3+1

        // d1+d2 (axis1) + d3+d4 (axis2) + d5+d6 (axis3)
        esum += c1 * (c2m1 - c2p1) + c2 * (c1p1 - c1m1)
              + c2 * (c0m2 - c0p2) + c0 * (c2p2 - c2m2)
              + c0 * (c1xm - c1xp) + c1 * (c0xp - c0xm);

        __syncthreads();  // protect LDS before next iteration's async writes
    }

    // Wave-level reduction via WMMA: D = ones(16x4) x B(4x16) + 0.
    // Every slot of B holds one lane's partial (v1 = 0), so row 0 of D summed
    // over its 16 lanes equals the exact wave total (pure f32 adds).
    v2f aones = {1.0f, 1.0f};
    v2f bdat  = {esum, 0.0f};
    v8f acc   = {};
    acc = __builtin_amdgcn_wmma_f32_16x16x4_f32(false, aones, false, bdat,
                                                (short)0, acc, false, false);
    float w = acc[0];
    w += __shfl_xor(w, 1, 32);
    w += __shfl_xor(w, 2, 32);
    w += __shfl_xor(w, 4, 32);
    w += __shfl_xor(w, 8, 32);   // lane 0 (and 16) now hold the wave total

    const int wid = t >> 5;
    if ((t & 31) == 0) wsum[wid] = w;
    __syncthreads();
    if (t == 0) {
        float s = 0.0f;
        for (int i = 0; i < 6; ++i) s += wsum[i];
        partial[blockIdx.y * gridDim.x + blockIdx.x] = s;
    }
}

__global__ __launch_bounds__(256) void dmi_reduce_kernel(
    const float* __restrict__ partial, int n,
    const float* __restrict__ D, float* __restrict__ out)
{
    __shared__ float red[256];
    float s = 0.0f;
    for (int i = threadIdx.x; i < n; i += 256) s += partial[i];
    red[threadIdx.x] = s;
    __syncthreads();
    for (int st = 128; st > 0; st >>= 1) {
        if ((int)threadIdx.x < st) red[threadIdx.x] += red[threadIdx.x + st];
        __syncthreads();
    }
    if (threadIdx.x == 0) {
        const float cells = (float)N * (float)N * (float)N;          // 192^3
        const float scale = 0.5f * D[0] * 4.0e-18f / cells;          // 0.5*D*CELLSIZE2/mean
        out[0] = red[0] * scale;
    }
}

extern "C" void kernel_launch(void* const* d_in, const int* in_sizes, int n_in,
                              void* d_out, int out_size, void* d_ws, size_t ws_size,
                              hipStream_t stream) {
    (void)in_sizes; (void)n_in; (void)out_size; (void)ws_size;
    const float* x = (const float*)d_in[0];   // (3,192,192,192) f32
    // d_in[1] = geo, unused by the reference math
    const float* D = (const float*)d_in[2];   // scalar
    float* out     = (float*)d_out;
    float* part    = (float*)d_ws;            // 9216 floats = 36 KB scratch

    dim3 grid(N / TILE, N);                   // (48, 192) blocks
    dmi_stencil_kernel<<<grid, 192, 0, stream>>>(x, part);
    dmi_reduce_kernel<<<1, 256, 0, stream>>>(part, (N / TILE) * N, D, out);
}